// Attention_82772609729141
// MI455X (gfx1250) — compile-verified
//
#include <hip/hip_runtime.h>
#include <hip/hip_bf16.h>
#include <stddef.h>

// ---------------------------------------------------------------------------
// Bilinear multi-head attention, MI455X (gfx1250, wave32, WMMA fp32 path).
//   B=8, L=2048, E=256, H=4, D=64
// Pipeline:
//   1) Wqb = Wq @ blockdiag(Wbil x H),  bqb = bq @ blockdiag(Wbil)   (tiny)
//   2) kx  = k @ Wk + bk          (WMMA f32 GEMM)
//   3) qw  = q @ Wqb + bqb        (WMMA f32 GEMM)
//   4) fused: S = qw kx^T, softmax, attn -> d_out (NT stores), ctx = P kx
//   5) out = ctx @ Wp + bp        (WMMA f32 GEMM)
// ---------------------------------------------------------------------------

typedef float v2f __attribute__((ext_vector_type(2)));
typedef float v8f __attribute__((ext_vector_type(8)));

static constexpr int Bc = 8;
static constexpr int Lc = 2048;
static constexpr int Ec = 256;
static constexpr int Hc = 4;
static constexpr int Dc = 64;
static constexpr int LDSTRIDE = Lc + 2;   // pad: rows land on distinct bank pairs

// ---------------------------------------------------------------------------
// Fold the per-head bilinear matrix into the q-projection weight/bias.
// Wqb[e, h*64+j] = sum_d Wq[e, h*64+d] * Wbil[d, j]
// ---------------------------------------------------------------------------
__global__ void prep_wqb_kernel(const float* __restrict__ Wq,
                                const float* __restrict__ Wbil,
                                float* __restrict__ Wqb) {
  int idx = blockIdx.x * blockDim.x + threadIdx.x;   // 256*256
  int e = idx >> 8;
  int c = idx & 255;
  int hh = c >> 6;
  int j  = c & 63;
  float acc = 0.0f;
#pragma unroll 8
  for (int d = 0; d < Dc; ++d)
    acc = fmaf(Wq[e * Ec + hh * Dc + d], Wbil[d * Dc + j], acc);
  Wqb[idx] = acc;
}

__global__ void prep_bqb_kernel(const float* __restrict__ bq,
                                const float* __restrict__ Wbil,
                                float* __restrict__ bqb) {
  int c = threadIdx.x;           // 256
  int hh = c >> 6;
  int j  = c & 63;
  float acc = 0.0f;
#pragma unroll 8
  for (int d = 0; d < Dc; ++d)
    acc = fmaf(bq[hh * Dc + d], Wbil[d * Dc + j], acc);
  bqb[c] = acc;
}

// ---------------------------------------------------------------------------
// Y[M,256] = X[M,256] @ W[256,256] + bias.  One 16x16 tile per wave,
// V_WMMA_F32_16X16X4_F32 over K=256.  256 threads = 8 waves per block.
// ---------------------------------------------------------------------------
__global__ __launch_bounds__(256) void gemm256_bias_kernel(
    const float* __restrict__ X, const float* __restrict__ W,
    const float* __restrict__ bias, float* __restrict__ Y) {
  const int wave = threadIdx.x >> 5;
  const int lane = threadIdx.x & 31;
  const int tile = blockIdx.x * 8 + wave;       // 16384 tiles total
  const int ct = tile & 15;                     // 16 column tiles
  const int rt = tile >> 4;                     // 1024 row tiles
  const int n  = lane & 15;                     // N (and A-row M)
  const int hi = lane >> 4;                     // lane half
  const int kb = hi * 2;                        // K sub-offset per ISA layout

  const float* xr = X + (size_t)(rt * 16 + n) * Ec;     // A row (M = lane&15)
  const float* wc = W + ct * 16 + n;                    // B column

  v8f c = {};
#pragma unroll 8
  for (int k = 0; k < Ec; k += 4) {
    v2f a = *(const v2f*)(xr + k + kb);                 // A[m][k+kb], [k+kb+1]
    v2f b;
    b.x = wc[(size_t)(k + kb) * Ec];                    // B[k+kb][n]
    b.y = wc[(size_t)(k + kb + 1) * Ec];                // B[k+kb+1][n]
    c = __builtin_amdgcn_wmma_f32_16x16x4_f32(false, a, false, b,
                                              (short)0, c, false, false);
  }
  const float bn = bias[ct * 16 + n];
  float* yr = Y + (size_t)(rt * 16) * Ec + ct * 16 + n;
#pragma unroll
  for (int r = 0; r < 8; ++r)
    yr[(size_t)(r + 8 * hi) * Ec] = c[r] + bn;          // D: M = r + 8*hi
}

// ---------------------------------------------------------------------------
// Fused score + softmax + attn streaming + PV.
// Block = 128 threads (4 waves) handles one (b, h, 16-row q tile).
// LDS: S[16][LDSTRIDE] f32 (~128 KB) + 8-float reduce scratch.
// ---------------------------------------------------------------------------
__global__ __launch_bounds__(128, 2) void attn_fused_kernel(
    const float* __restrict__ qw, const float* __restrict__ kx,
    float* __restrict__ attn, float* __restrict__ ctx) {
  extern __shared__ float smem[];
  float* Sl  = smem;                       // 16 x LDSTRIDE
  float* red = smem + 16 * LDSTRIDE;       // 8 floats

  const int qt = blockIdx.x & 127;         // q tile (128 per (b,h))
  const int bh = blockIdx.x >> 7;
  const int b  = bh & (Bc - 1);
  const int h  = bh >> 3;
  const int q0 = qt * 16;

  const int tid  = threadIdx.x;
  const int wave = tid >> 5;
  const int lane = tid & 31;
  const int n    = lane & 15;              // N index / A-row M
  const int hi   = lane >> 4;
  const int kb   = hi * 2;

  const float* qwb = qw + (size_t)(b * Lc + q0) * Ec + h * Dc;  // [16 x 64], ld=256
  const float* kxb = kx + (size_t)b * Lc * Ec + h * Dc;         // [2048 x 64], ld=256

  // ---- Phase 1: S[16][2048] = qw_tile @ kx^T  (K = D = 64) ----
  v2f afrag[16];
#pragma unroll
  for (int kc = 0; kc < 16; ++kc)
    afrag[kc] = *(const v2f*)(qwb + n * Ec + kc * 4 + kb);      // A[m][4kc+kb..]

  for (int kt = wave; kt < 128; kt += 4) {                      // key tiles of 16
    v8f c = {};
    const float* kr = kxb + (size_t)(kt * 16 + n) * Ec;         // B col n = key row
#pragma unroll
    for (int kc = 0; kc < 16; ++kc) {
      v2f bfr = *(const v2f*)(kr + kc * 4 + kb);                // B[4kc+kb][n]..
      c = __builtin_amdgcn_wmma_f32_16x16x4_f32(false, afrag[kc], false, bfr,
                                                (short)0, c, false, false);
    }
#pragma unroll
    for (int r = 0; r < 8; ++r)
      Sl[(r + 8 * hi) * LDSTRIDE + kt * 16 + n] = c[r];
  }
  __syncthreads();

  // ---- Phase 2: row softmax; stream attn with non-temporal stores ----
  float* attnb = attn + ((size_t)((h * Bc + b) * Lc + q0)) * Lc;
  for (int row = 0; row < 16; ++row) {
    float* Sr = Sl + row * LDSTRIDE;
    float m_loc = -3.402823466e38f;
#pragma unroll
    for (int i = 0; i < 16; ++i)
      m_loc = fmaxf(m_loc, Sr[tid + 128 * i]);
#pragma unroll
    for (int off = 16; off > 0; off >>= 1)
      m_loc = fmaxf(m_loc, __shfl_xor(m_loc, off, 32));
    if (lane == 0) red[wave] = m_loc;
    __syncthreads();
    const float mrow = fmaxf(fmaxf(red[0], red[1]), fmaxf(red[2], red[3]));
    __syncthreads();

    float s_loc = 0.0f;
#pragma unroll
    for (int i = 0; i < 16; ++i) {
      float e = __expf(Sr[tid + 128 * i] - mrow);
      Sr[tid + 128 * i] = e;
      s_loc += e;
    }
#pragma unroll
    for (int off = 16; off > 0; off >>= 1)
      s_loc += __shfl_xor(s_loc, off, 32);
    if (lane == 0) red[wave] = s_loc;
    __syncthreads();
    const float rden = 1.0f / (red[0] + red[1] + red[2] + red[3]);
    __syncthreads();

    float* arow = attnb + (size_t)row * Lc;
#pragma unroll
    for (int i = 0; i < 16; ++i) {
      float p = Sr[tid + 128 * i] * rden;
      Sr[tid + 128 * i] = p;
      __builtin_nontemporal_store(p, arow + tid + 128 * i);    // attn never re-read
    }
  }
  __syncthreads();

  // ---- Phase 3: ctx[16][64] = P[16][2048] @ kx[2048][64]; one 16-col tile/wave ----
  {
    const int col0 = wave * 16;
    const float* kcol = kxb + col0;                            // B cols = ctx dims
    v8f c = {};
#pragma unroll 4
    for (int kk = 0; kk < 512; ++kk) {                         // K = 2048 / 4
      const int k0 = kk * 4 + kb;
      v2f a = *(const v2f*)(Sl + n * LDSTRIDE + k0);           // A[m][k0], [k0+1]
      v2f bfr;
      bfr.x = kcol[(size_t)k0 * Ec + n];                       // B[k0][n]
      bfr.y = kcol[(size_t)(k0 + 1) * Ec + n];                 // B[k0+1][n]
      c = __builtin_amdgcn_wmma_f32_16x16x4_f32(false, a, false, bfr,
                                                (short)0, c, false, false);
    }
    float* ctxb = ctx + (size_t)(b * Lc + q0) * Ec + h * Dc + col0;
#pragma unroll
    for (int r = 0; r < 8; ++r)
      ctxb[(size_t)(r + 8 * hi) * Ec + n] = c[r];
  }
}

// ---------------------------------------------------------------------------
extern "C" void kernel_launch(void* const* d_in, const int* in_sizes, int n_in,
                              void* d_out, int out_size, void* d_ws, size_t ws_size,
                              hipStream_t stream) {
  const float* k    = (const float*)d_in[0];
  const float* q    = (const float*)d_in[1];
  const float* Wk   = (const float*)d_in[2];
  const float* bk   = (const float*)d_in[3];
  const float* Wq   = (const float*)d_in[4];
  const float* bq   = (const float*)d_in[5];
  const float* Wbil = (const float*)d_in[6];
  const float* Wp   = (const float*)d_in[7];
  const float* bp   = (const float*)d_in[8];

  const size_t BLE = (size_t)Bc * Lc * Ec;      // 4,194,304

  float* out  = (float*)d_out;                  // [B, L, E]
  float* attn = out + BLE;                      // [H*B, L, L]

  float* ws  = (float*)d_ws;
  float* kx  = ws;                              // [B, L, E]
  float* qw  = kx + BLE;                        // [B, L, E]
  float* ctx = qw + BLE;                        // [B, L, E]
  float* Wqb = ctx + BLE;                       // [E, E]
  float* bqb = Wqb + (size_t)Ec * Ec;           // [E]

  // 1) fold bilinear into q projection
  prep_wqb_kernel<<<256, 256, 0, stream>>>(Wq, Wbil, Wqb);
  prep_bqb_kernel<<<1, 256, 0, stream>>>(bq, Wbil, bqb);

  // 2/3) projections: 16384x256 @ 256x256 ; 16384 wave-tiles / 8 waves per block
  gemm256_bias_kernel<<<2048, 256, 0, stream>>>(k, Wk, bk, kx);
  gemm256_bias_kernel<<<2048, 256, 0, stream>>>(q, Wqb, bqb, qw);

  // 4) fused attention: B*H*(L/16) = 4096 workgroups of 128 threads
  const size_t smem_bytes = (size_t)(16 * LDSTRIDE + 8) * sizeof(float);
  attn_fused_kernel<<<4096, 128, smem_bytes, stream>>>(qw, kx, attn, ctx);

  // 5) output projection
  gemm256_bias_kernel<<<2048, 256, 0, stream>>>(ctx, Wp, bp, out);
}